// FasterRCNN_74423193305783
// MI455X (gfx1250) — compile-verified
//
#include <hip/hip_runtime.h>
#include <hip/hip_bf16.h>
#include <stdint.h>

#define NB 16          // batch
#define NP 4096        // proposals per image
#define NCLS 20
#define KOUT 100
#define SCORE_THRESH 0.05f
#define NMS_THRESH 0.5f
#define MIN_SIZE 0.01f
#define BBOX_XFORM_CLIP 4.135166556742356f  // log(1000/16)

// ---- CDNA5 async global->LDS staging (ASYNCcnt-tracked) ----
__device__ __forceinline__ void async_load_b128(uint32_t lds_off, const void* gaddr) {
    // GV mode: vdst = LDS byte offset VGPR, vaddr = 64-bit global address pair
    asm volatile("global_load_async_to_lds_b128 %0, %1, off"
                 :: "v"(lds_off), "v"(gaddr) : "memory");
}
__device__ __forceinline__ void wait_async0() {
    asm volatile("s_wait_asynccnt 0" ::: "memory");
}

// ---------------- Kernel 1: decode + dual softmax + score/label/valid ----------------
__global__ __launch_bounds__(256) void decode_score_kernel(
        const float* __restrict__ logits,   // [NB*NP, 20]
        const float* __restrict__ reg,      // [NB*NP, 80]
        const float* __restrict__ props,    // [NB*NP, 4]
        const int*   __restrict__ imgsz,    // [1]
        float* __restrict__ boxes,          // [NB*NP,4] clipped
        float* __restrict__ scores,         // [NB*NP] (-1 if invalid)
        int*   __restrict__ labels)         // [NB*NP]
{
    int n = blockIdx.x * blockDim.x + threadIdx.x;
    if (n >= NB * NP) return;
    float img = (float)imgsz[0];

    float lv[NCLS];
    const float* l = logits + (size_t)n * NCLS;
#pragma unroll
    for (int i = 0; i < NCLS; i += 4) {
        float4 t = *(const float4*)(l + i);
        lv[i] = t.x; lv[i+1] = t.y; lv[i+2] = t.z; lv[i+3] = t.w;
    }
    // softmax group A: classes 0..14
    float mA = lv[0];
#pragma unroll
    for (int i = 1; i < 15; i++) mA = fmaxf(mA, lv[i]);
    float dA = 0.f;
#pragma unroll
    for (int i = 0; i < 15; i++) dA += expf(lv[i] - mA);
    int vj = 1; float vb = lv[1];
#pragma unroll
    for (int i = 2; i < 15; i++) if (lv[i] > vb) { vb = lv[i]; vj = i; }  // first-max like argmax
    float val_s = expf(vb - mA) / dA;
    // softmax group B: classes 15..19
    float mB = lv[15];
#pragma unroll
    for (int i = 16; i < 20; i++) mB = fmaxf(mB, lv[i]);
    float dB = 0.f;
#pragma unroll
    for (int i = 15; i < 20; i++) dB += expf(lv[i] - mB);
    int cj = 16; float cb = lv[16];
#pragma unroll
    for (int i = 17; i < 20; i++) if (lv[i] > cb) { cb = lv[i]; cj = i; }
    float col_s = expf(cb - mB) / dB;

    float score = 0.5f * (val_s + col_s);
    int label = vj * 10 + (cj - 15);   // (values+1)*10 + (colors+1)

    // decode with class-1 deltas (columns 4..7), weights (10,10,5,5)
    float4 pr = *(const float4*)(props + (size_t)n * 4);
    float4 rl = *(const float4*)(reg + (size_t)n * (NCLS * 4) + 4);
    float w = pr.z - pr.x, h = pr.w - pr.y;
    float cx = pr.x + 0.5f * w, cy = pr.y + 0.5f * h;
    float dx = rl.x * 0.1f, dy = rl.y * 0.1f;
    float dw = fminf(rl.z * 0.2f, BBOX_XFORM_CLIP);
    float dh = fminf(rl.w * 0.2f, BBOX_XFORM_CLIP);
    float pcx = dx * w + cx, pcy = dy * h + cy;
    float pw = expf(dw) * w, ph = expf(dh) * h;
    float x1 = pcx - 0.5f * pw, y1 = pcy - 0.5f * ph;
    float x2 = pcx + 0.5f * pw, y2 = pcy + 0.5f * ph;
    x1 = fminf(fmaxf(x1, 0.f), img); y1 = fminf(fmaxf(y1, 0.f), img);
    x2 = fminf(fmaxf(x2, 0.f), img); y2 = fminf(fmaxf(y2, 0.f), img);
    float bw = x2 - x1, bh = y2 - y1;
    bool valid = (score > SCORE_THRESH) && (bw >= MIN_SIZE) && (bh >= MIN_SIZE);

    ((float4*)boxes)[n] = make_float4(x1, y1, x2, y2);
    scores[n] = valid ? score : -1.0f;
    labels[n] = label;
}

// ---------------- Kernel 2: per-image bitonic sort (desc score, asc index) + gather ----------------
__global__ __launch_bounds__(1024) void sort_kernel(
        const float* __restrict__ boxes, const float* __restrict__ scores,
        const int* __restrict__ labels, const int* __restrict__ imgsz,
        float* __restrict__ sbox,    // sorted clipped boxes      [NB*NP*4]
        float* __restrict__ snb,     // sorted label-offset boxes [NB*NP*4]
        float* __restrict__ sscore,  // sorted effective scores   [NB*NP]
        int*   __restrict__ slabel)  // sorted labels             [NB*NP]
{
    __shared__ float ss[NP];
    __shared__ int   si[NP];
    const int b = blockIdx.x, t = threadIdx.x, T = blockDim.x;
    for (int r = t; r < NP; r += T) { ss[r] = scores[b * NP + r]; si[r] = r; }
    for (int k = 2; k <= NP; k <<= 1) {
        for (int j = k >> 1; j > 0; j >>= 1) {
            __syncthreads();
            for (int i = t; i < NP; i += T) {
                int x = i ^ j;
                if (x > i) {
                    float a = ss[i], c = ss[x];
                    int  ai = si[i], ci = si[x];
                    bool before = (a > c) || (a == c && ai < ci);  // desired (descending) order
                    bool up = ((i & k) == 0);
                    if (up ? !before : before) { ss[i] = c; ss[x] = a; si[i] = ci; si[x] = ai; }
                }
            }
        }
    }
    __syncthreads();
    float off_scale = (float)imgsz[0] + 1.0f;
    for (int r = t; r < NP; r += T) {
        int p = si[r];
        float4 bx = ((const float4*)boxes)[b * NP + p];
        int lab = labels[b * NP + p];
        float off = (float)lab * off_scale;
        ((float4*)sbox)[b * NP + r] = bx;
        ((float4*)snb)[b * NP + r] = make_float4(bx.x + off, bx.y + off, bx.z + off, bx.w + off);
        sscore[b * NP + r] = ss[r];
        slabel[b * NP + r] = lab;
    }
}

// ---------------- Kernel 3: LDS-resident greedy NMS + compaction + top-K output ----------------
__global__ __launch_bounds__(1024) void nms_kernel(
        const float* __restrict__ sbox, const float* __restrict__ snb,
        const float* __restrict__ sscore, const int* __restrict__ slabel,
        float* __restrict__ out)
{
    extern __shared__ char smem[];
    float4* nb  = (float4*)smem;                           // NP*16 = 64 KB
    float*  sc  = (float*)(smem + NP * 16);                // NP*4  = 16 KB
    int*    sup = (int*)  (smem + NP * 16 + NP * 4);       // NP*4  = 16 KB
    int*    scn = (int*)  (smem + NP * 16 + NP * 8);       // 1024*4 = 4 KB
    const int b = blockIdx.x, t = threadIdx.x;
    const int T = 1024;

    float* ob  = out + (size_t)b * KOUT * 4;
    float* osc = out + (size_t)NB * KOUT * 4 + (size_t)b * KOUT;
    float* olb = out + (size_t)NB * KOUT * 4 + (size_t)NB * KOUT + (size_t)b * KOUT;
    if (t < KOUT * 4) ob[t] = 0.f;
    if (t < KOUT) { osc[t] = 0.f; olb[t] = 0.f; }

    // async-stage sorted offset-boxes + scores into LDS (thread t owns ranks 4t..4t+3)
    const float4* gnb = ((const float4*)snb) + (size_t)b * NP;
    const float*  gsc = sscore + (size_t)b * NP;
#pragma unroll
    for (int q = 0; q < 4; q++) {
        int r = t * 4 + q;
        async_load_b128((uint32_t)(uintptr_t)&nb[r], (const void*)&gnb[r]);
        sup[r] = 0;
    }
    async_load_b128((uint32_t)(uintptr_t)&sc[t * 4], (const void*)&gsc[t * 4]);
    wait_async0();
    __syncthreads();

    // Greedy suppression. keep_i / kept / breaks are block-uniform (identical
    // computation from shared state), so conditional barriers execute uniformly.
    // Early exits (both exact w.r.t. the reference output):
    //  - scores sorted descending, invalid == -1  => once sc[i] <= 0, stop.
    //  - only the first KOUT kept boxes are output, and a kept box beyond the
    //    KOUT-th can only suppress later ranks (never output) => stop at KOUT,
    //    remembering i_end so compaction ignores ranks beyond it.
    int kept = 0;
    int iend = NP - 1;
    for (int i = 0; i < NP; i++) {
        float si_sc = sc[i];
        if (si_sc <= 0.f) { iend = i; break; }     // uniform: all later ranks invalid
        bool keep_i = (sup[i] == 0);
        if (keep_i) {
            kept++;
            if (kept >= KOUT) { iend = i; break; } // uniform: sup[] final for r <= iend
            float4 bi = nb[i];
            float ai = (bi.z - bi.x) * (bi.w - bi.y);
#pragma unroll
            for (int q = 0; q < 4; q++) {
                int r = t * 4 + q;
                if (r > i && sup[r] == 0) {
                    float4 bj = nb[r];
                    float aj = (bj.z - bj.x) * (bj.w - bj.y);
                    float ix = fmaxf(fminf(bi.z, bj.z) - fmaxf(bi.x, bj.x), 0.f);
                    float iy = fmaxf(fminf(bi.w, bj.w) - fmaxf(bi.y, bj.y), 0.f);
                    float inter = ix * iy;
                    float uni = ai + aj - inter;
                    float iou = (uni > 0.f) ? inter / uni : 0.f;
                    if (iou > NMS_THRESH) sup[r] = 1;   // owner-exclusive write: race-free
                }
            }
            __syncthreads();
        }
    }
    __syncthreads();

    // compaction over ranks [0, iend]: per-thread counts + Hillis-Steele block scan
    int kflag[4]; int cnt = 0;
#pragma unroll
    for (int q = 0; q < 4; q++) {
        int r = t * 4 + q;
        kflag[q] = (r <= iend && sc[r] > 0.f && sup[r] == 0) ? 1 : 0;
        cnt += kflag[q];
    }
    scn[t] = cnt;
    __syncthreads();
    for (int d = 1; d < T; d <<= 1) {
        int v = (t >= d) ? scn[t - d] : 0;
        __syncthreads();
        scn[t] += v;
        __syncthreads();
    }
    int pos = scn[t] - cnt;  // exclusive prefix for this thread's first owned rank
#pragma unroll
    for (int q = 0; q < 4; q++) {
        int r = t * 4 + q;
        if (kflag[q]) {
            if (pos < KOUT) {
                float4 bx = ((const float4*)sbox)[b * NP + r];
                ob[pos * 4 + 0] = bx.x; ob[pos * 4 + 1] = bx.y;
                ob[pos * 4 + 2] = bx.z; ob[pos * 4 + 3] = bx.w;
                osc[pos] = sc[r];
                olb[pos] = (float)slabel[b * NP + r];
            }
            pos++;
        }
    }
}

extern "C" void kernel_launch(void* const* d_in, const int* in_sizes, int n_in,
                              void* d_out, int out_size, void* d_ws, size_t ws_size,
                              hipStream_t stream) {
    const float* logits = (const float*)d_in[0];   // [NB*NP,20]
    const float* reg    = (const float*)d_in[1];   // [NB*NP,80]
    const float* props  = (const float*)d_in[2];   // [NB,NP,4]
    const int*   imgsz  = (const int*)d_in[3];     // [1]
    float* out = (float*)d_out;                    // boxes[NB,K,4] ++ scores[NB,K] ++ labels[NB,K]

    // workspace layout (floats): 4 MiB total
    float* ws = (float*)d_ws;
    const size_t BP = (size_t)NB * NP;
    float* boxes  = ws;                       // BP*4
    float* scores = boxes + BP * 4;           // BP
    int*   labels = (int*)(scores + BP);      // BP
    float* sbox   = (float*)(labels + BP);    // BP*4
    float* snb    = sbox + BP * 4;            // BP*4
    float* sscore = snb + BP * 4;             // BP
    int*   slabel = (int*)(sscore + BP);      // BP

    decode_score_kernel<<<(NB * NP + 255) / 256, 256, 0, stream>>>(
        logits, reg, props, imgsz, boxes, scores, labels);
    sort_kernel<<<NB, 1024, 0, stream>>>(
        boxes, scores, labels, imgsz, sbox, snb, sscore, slabel);
    size_t smem = (size_t)NP * 16 + NP * 4 + NP * 4 + 1024 * 4;  // 100 KB dynamic LDS
    nms_kernel<<<NB, 1024, smem, stream>>>(sbox, snb, sscore, slabel, out);
}